// STEncoder_85942295593054
// MI455X (gfx1250) — compile-verified
//
#include <hip/hip_runtime.h>
#include <math.h>

typedef float v2f __attribute__((ext_vector_type(2)));
typedef float v8f __attribute__((ext_vector_type(8)));
typedef unsigned int u32x4 __attribute__((ext_vector_type(4)));
typedef int i32x8 __attribute__((ext_vector_type(8)));
typedef int i32x4 __attribute__((ext_vector_type(4)));

#define BATCH 16
#define NNODE 2000
#define TIN0  12
#define DIN   3
#define DOUT  64
#define CMID  32

#define KTILE 128   // K-rows of B staged per TDM transfer
#define NTILE 64    // N-columns per block tile

// Issue a TDM transfer of a 2D fp32 tile [tile_rows x NTILE] from global
// (row stride = Nc elements) into LDS at lds_addr.  D# per CDNA5 ISA ch.8.
// Tensor dims are given relative to the tile start so trailing N/K tiles
// zero-fill out-of-bounds elements.
__device__ __forceinline__ void tdm_load_B(const float* gptr, unsigned lds_addr,
                                           unsigned td0, unsigned td1,
                                           int tile_rows, int Nc) {
  unsigned long long gaddr = (unsigned long long)(const void*)gptr;
  u32x4 g0;
  g0[0] = (unsigned)__builtin_amdgcn_readfirstlane(1);                 // count=1
  g0[1] = (unsigned)__builtin_amdgcn_readfirstlane((int)lds_addr);
  g0[2] = (unsigned)__builtin_amdgcn_readfirstlane((int)(gaddr & 0xFFFFFFFFull));
  g0[3] = (unsigned)__builtin_amdgcn_readfirstlane(
              (int)(((gaddr >> 32) & 0x1FFFFFFull) | (2u << 30)));     // type=2
  i32x8 g1;
  g1[0] = __builtin_amdgcn_readfirstlane((int)(2u << 16));             // data_size=4B
  g1[1] = __builtin_amdgcn_readfirstlane((int)((td0 & 0xFFFFu) << 16));
  g1[2] = __builtin_amdgcn_readfirstlane(
              (int)(((td0 >> 16) & 0xFFFFu) | ((td1 & 0xFFFFu) << 16)));
  g1[3] = __builtin_amdgcn_readfirstlane(
              (int)(((td1 >> 16) & 0xFFFFu) | ((unsigned)NTILE << 16))); // tile_dim0
  g1[4] = __builtin_amdgcn_readfirstlane((int)(unsigned)tile_rows);    // tile_dim1
  g1[5] = __builtin_amdgcn_readfirstlane((int)(unsigned)Nc);           // dim0 stride lo
  g1[6] = __builtin_amdgcn_readfirstlane(0);                           // stride hi
  g1[7] = __builtin_amdgcn_readfirstlane(0);
  i32x4 g2 = {0, 0, 0, 0};
  i32x4 g3 = {0, 0, 0, 0};
  i32x8 g4 = {0, 0, 0, 0, 0, 0, 0, 0};
  __builtin_amdgcn_tensor_load_to_lds(g0, g1, g2, g3, g4, 0);
}

// ---------------------------------------------------------------------------
// fp32 WMMA GEMM: C[M,Nc] = A[M,K] @ B[K,Nc], row-major, lda=K, ldb=ldc=Nc.
// 4 waves/block; block tile = 256(M) x 64(N); each wave owns 64(M) x 64(N)
// = 4x4 fragments of V_WMMA_F32_16X16X4_F32 (16 accumulators).
// B tiles [KTILE x 64] are double-buffered in LDS via the Tensor Data Mover:
// wave 0 issues TENSOR_LOAD_TO_LDS for tile t+1, all waves compute tile t,
// then s_wait_tensorcnt + barrier rotate the buffers.  A fragments are
// software-pipelined by one K-step to hide global load latency.
// Fragment layouts per CDNA5 ISA 7.12.2:
//   A (16x4 f32): lanes 0-15 -> M=lane, {K=0,K=1}; lanes 16-31 -> {K=2,K=3}
//   B (4x16 f32): lanes 0-15 -> N=lane, {K=0,K=1}; lanes 16-31 -> {K=2,K=3}
//   D (16x16 f32): VGPR v: lanes0-15 M=v, lanes16-31 M=v+8; N=lane%16
// ---------------------------------------------------------------------------
__global__ __launch_bounds__(128) void gemm_wmma_f32(
    const float* __restrict__ A, const float* __restrict__ Bm,
    float* __restrict__ C, int M, int K, int Nc) {
  __shared__ float smemB[2][KTILE * NTILE];   // 2 x 32 KB

  const int lane = threadIdx.x & 31;
  const int wave = threadIdx.x >> 5;
  const int half = lane >> 4;
  const int l16  = lane & 15;
  const int m_base = blockIdx.y * 256 + wave * 64;
  const int n_base = blockIdx.x * NTILE;

  v8f acc[4][4] = {};

  const float* aRow[4];
#pragma unroll
  for (int i = 0; i < 4; ++i) {
    int r = m_base + 16 * i + l16;
    r = (r < M) ? r : (M - 1);            // clamp; invalid tiles discarded at store
    aRow[i] = A + (size_t)r * K + 2 * half;
  }

  const int ntiles = (K + KTILE - 1) / KTILE;

  // prologue: stage tile 0 into buffer 0
  if (wave == 0) {
    tdm_load_B(Bm + n_base, (unsigned)(unsigned long long)(&smemB[0][0]),
               (unsigned)(Nc - n_base), (unsigned)K,
               (K < KTILE) ? K : KTILE, Nc);
    __builtin_amdgcn_s_wait_tensorcnt(0);
  }
  __syncthreads();

  for (int tt = 0; tt < ntiles; ++tt) {
    const int kk0 = tt * KTILE;
    const int kt = (K - kk0 < KTILE) ? (K - kk0) : KTILE;
    const float* bufc = &smemB[tt & 1][0];

    // kick off TDM for the next tile into the other buffer (overlaps compute)
    if (wave == 0 && tt + 1 < ntiles) {
      const int kn0 = kk0 + KTILE;
      tdm_load_B(Bm + (size_t)kn0 * Nc + n_base,
                 (unsigned)(unsigned long long)(&smemB[(tt + 1) & 1][0]),
                 (unsigned)(Nc - n_base), (unsigned)(K - kn0),
                 (K - kn0 < KTILE) ? (K - kn0) : KTILE, Nc);
    }

    // software-pipelined K loop: af holds fragments for step kk, af_n prefetches kk+4
    v2f af[4];
#pragma unroll
    for (int i = 0; i < 4; ++i) {
      const float* p = aRow[i] + kk0;
      af[i][0] = p[0];
      af[i][1] = p[1];
    }
    for (int kk = 0; kk < kt; kk += 4) {
      v2f bf[4];
#pragma unroll
      for (int j = 0; j < 4; ++j) {
        const float* s = bufc + (kk + 2 * half) * NTILE + 16 * j + l16;
        bf[j][0] = s[0];
        bf[j][1] = s[NTILE];
      }
      const int kkn = (kk + 4 < kt) ? (kk + 4) : kk;   // clamped prefetch index
      v2f af_n[4];
#pragma unroll
      for (int i = 0; i < 4; ++i) {
        const float* p = aRow[i] + kk0 + kkn;
        af_n[i][0] = p[0];
        af_n[i][1] = p[1];
      }
#pragma unroll
      for (int i = 0; i < 4; ++i)
#pragma unroll
        for (int j = 0; j < 4; ++j)
          acc[i][j] = __builtin_amdgcn_wmma_f32_16x16x4_f32(
              false, af[i], false, bf[j], (short)0, acc[i][j], false, false);
#pragma unroll
      for (int i = 0; i < 4; ++i) af[i] = af_n[i];
    }

    if (wave == 0) __builtin_amdgcn_s_wait_tensorcnt(0);  // next tile landed
    __syncthreads();                                      // rotate buffers
  }

#pragma unroll
  for (int i = 0; i < 4; ++i) {
    int m_sub = m_base + 16 * i;
    if (m_sub >= M) continue;
#pragma unroll
    for (int j = 0; j < 4; ++j) {
      int n_sub = n_base + 16 * j;
      if (n_sub >= Nc) continue;
#pragma unroll
      for (int v = 0; v < 8; ++v) {
        int row = m_sub + v + 8 * half;
        C[(size_t)row * Nc + n_sub + l16] = acc[i][j][v];
      }
    }
  }
}

// ---------------------------------------------------------------------------
// d[n] = clip(rowsum(graph)[n] + 1, 1e-6)^-0.5
// ---------------------------------------------------------------------------
__global__ void degree_kernel(const float* __restrict__ graph, float* __restrict__ d) {
  const int n = blockIdx.x;
  __shared__ float s[256];
  float sum = 0.f;
  for (int m = threadIdx.x; m < NNODE; m += 256) sum += graph[(size_t)n * NNODE + m];
  s[threadIdx.x] = sum;
  __syncthreads();
  for (int st = 128; st > 0; st >>= 1) {
    if ((int)threadIdx.x < st) s[threadIdx.x] += s[threadIdx.x + st];
    __syncthreads();
  }
  if (threadIdx.x == 0) {
    float g = s[0] + 1.0f;                    // + diagonal of (graph + I)
    g = (g < 1e-6f) ? 1e-6f : g;
    d[n] = rsqrtf(g);
  }
}

// lapT[m][n] = lap[n][m] = delta(n,m) - d[n]*(graph[n][m] + delta)*d[m]
__global__ void lapT_kernel(const float* __restrict__ graph, const float* __restrict__ d,
                            float* __restrict__ lapT) {
  size_t idx = (size_t)blockIdx.x * blockDim.x + threadIdx.x;
  if (idx >= (size_t)NNODE * NNODE) return;
  int m = (int)(idx / NNODE);
  int n = (int)(idx % NNODE);
  float diag = (m == n) ? 1.0f : 0.0f;
  float g = graph[(size_t)n * NNODE + m] + diag;
  lapT[idx] = diag - d[n] * g * d[m];
}

// cheb2T = 2*(lapT@lapT) - I   (in-place fixup after GEMM)
__global__ void cheb2_fix_kernel(float* __restrict__ c2) {
  size_t idx = (size_t)blockIdx.x * blockDim.x + threadIdx.x;
  if (idx >= (size_t)NNODE * NNODE) return;
  int m = (int)(idx / NNODE);
  int n = (int)(idx % NNODE);
  c2[idx] = 2.0f * c2[idx] - ((m == n) ? 1.0f : 0.0f);
}

// h[b,o,t,n] = sum_c x[b,c,t,n]*in_w[c,o] + in_b[o]
__global__ void input_conv_kernel(const float* __restrict__ x, const float* __restrict__ w,
                                  const float* __restrict__ bias, float* __restrict__ h) {
  const size_t total = (size_t)BATCH * DOUT * TIN0 * NNODE;
  size_t idx = (size_t)blockIdx.x * blockDim.x + threadIdx.x;
  if (idx >= total) return;
  int n = (int)(idx % NNODE);
  int t = (int)((idx / NNODE) % TIN0);
  int o = (int)((idx / ((size_t)NNODE * TIN0)) % DOUT);
  int b = (int)(idx / ((size_t)NNODE * TIN0 * DOUT));
  float acc = bias[o];
#pragma unroll
  for (int c = 0; c < DIN; ++c)
    acc += x[(((size_t)b * DIN + c) * TIN0 + t) * NNODE + n] * w[c * DOUT + o];
  h[idx] = acc;
}

// GLU temporal conv: out[b,i,t,n] = (conv_p + align) * sigmoid(conv_q)
__global__ void glu_kernel(const float* __restrict__ h, const float* __restrict__ w,
                           const float* __restrict__ bias, const float* __restrict__ aw,
                           const float* __restrict__ ab, float* __restrict__ out, int Tin) {
  const int Tout = Tin - 2;
  const size_t total = (size_t)BATCH * CMID * Tout * NNODE;
  size_t idx = (size_t)blockIdx.x * blockDim.x + threadIdx.x;
  if (idx >= total) return;
  int n = (int)(idx % NNODE);
  int t = (int)((idx / NNODE) % Tout);
  int i = (int)((idx / ((size_t)NNODE * Tout)) % CMID);
  int b = (int)(idx / ((size_t)NNODE * Tout * CMID));
  float p = bias[i];
  float q = bias[i + CMID];
  float a = ab[i];
  for (int c = 0; c < DOUT; ++c) {
    const float* hp = h + (((size_t)b * DOUT + c) * Tin + t) * NNODE + n;
    float h0 = hp[0], h1 = hp[NNODE], h2 = hp[2 * NNODE];
    const float* wp = w + ((size_t)i * DOUT + c) * 3;
    const float* wq = w + ((size_t)(i + CMID) * DOUT + c) * 3;
    p += h0 * wp[0] + h1 * wp[1] + h2 * wp[2];
    q += h0 * wq[0] + h1 * wq[1] + h2 * wq[2];
    a += h2 * aw[i * DOUT + c];
  }
  out[idx] = (p + a) * (1.0f / (1.0f + __expf(-q)));
}

// sconv channel mix: out = relu( sum_i th[i,o,0]*g + th[i,o,1]*z1 + th[i,o,2]*z2 + bias + g_resid )
__global__ void sconv_kernel(const float* __restrict__ g, const float* __restrict__ z1,
                             const float* __restrict__ z2, const float* __restrict__ th,
                             const float* __restrict__ bias, float* __restrict__ out, int T) {
  const size_t total = (size_t)BATCH * CMID * T * NNODE;
  size_t idx = (size_t)blockIdx.x * blockDim.x + threadIdx.x;
  if (idx >= total) return;
  int n = (int)(idx % NNODE);
  int t = (int)((idx / NNODE) % T);
  int o = (int)((idx / ((size_t)NNODE * T)) % CMID);
  int b = (int)(idx / ((size_t)NNODE * T * CMID));
  float acc = bias[o] + g[idx];                 // residual (identity align)
  for (int i = 0; i < CMID; ++i) {
    size_t r = (((size_t)b * CMID + i) * T + t) * NNODE + n;
    const float* tp = th + ((size_t)i * CMID + o) * 3;
    acc += tp[0] * g[r] + tp[1] * z1[r] + tp[2] * z2[r];
  }
  out[idx] = acc > 0.f ? acc : 0.f;
}

// ReLU temporal conv with channel-padded residual
__global__ void relu_tconv_kernel(const float* __restrict__ s, const float* __restrict__ w,
                                  const float* __restrict__ bias, float* __restrict__ out,
                                  int Tin) {
  const int Tout = Tin - 2;
  const size_t total = (size_t)BATCH * DOUT * Tout * NNODE;
  size_t idx = (size_t)blockIdx.x * blockDim.x + threadIdx.x;
  if (idx >= total) return;
  int n = (int)(idx % NNODE);
  int t = (int)((idx / NNODE) % Tout);
  int o = (int)((idx / ((size_t)NNODE * Tout)) % DOUT);
  int b = (int)(idx / ((size_t)NNODE * Tout * DOUT));
  float acc = bias[o];
  for (int i = 0; i < CMID; ++i) {
    const float* sp = s + (((size_t)b * CMID + i) * Tin + t) * NNODE + n;
    const float* wp = w + ((size_t)o * CMID + i) * 3;
    acc += sp[0] * wp[0] + sp[NNODE] * wp[1] + sp[2 * NNODE] * wp[2];
  }
  if (o < CMID) acc += s[(((size_t)b * CMID + o) * Tin + t + 2) * NNODE + n];
  out[idx] = acc > 0.f ? acc : 0.f;
}

// LayerNorm over (N, C) per (b, t); gamma/beta shaped [N, C]
__global__ void layernorm_kernel(const float* __restrict__ x, const float* __restrict__ gam,
                                 const float* __restrict__ bet, float* __restrict__ y, int T) {
  const int bt = blockIdx.x;
  const int b = bt / T;
  const int t = bt % T;
  __shared__ float s_sum[256];
  __shared__ float s_sq[256];
  const int total = DOUT * NNODE;
  float sum = 0.f, sq = 0.f;
  for (int idx = threadIdx.x; idx < total; idx += 256) {
    int c = idx / NNODE, n = idx % NNODE;
    float v = x[(((size_t)b * DOUT + c) * T + t) * NNODE + n];
    sum += v;
    sq += v * v;
  }
  s_sum[threadIdx.x] = sum;
  s_sq[threadIdx.x] = sq;
  __syncthreads();
  for (int st = 128; st > 0; st >>= 1) {
    if ((int)threadIdx.x < st) {
      s_sum[threadIdx.x] += s_sum[threadIdx.x + st];
      s_sq[threadIdx.x] += s_sq[threadIdx.x + st];
    }
    __syncthreads();
  }
  float mu = s_sum[0] / (float)total;
  float var = s_sq[0] / (float)total - mu * mu;
  float inv = rsqrtf(var + 1e-5f);
  for (int idx = threadIdx.x; idx < total; idx += 256) {
    int c = idx / NNODE, n = idx % NNODE;
    size_t off = (((size_t)b * DOUT + c) * T + t) * NNODE + n;
    y[off] = (x[off] - mu) * inv * gam[(size_t)n * DOUT + c] + bet[(size_t)n * DOUT + c];
  }
}

// ---------------------------------------------------------------------------
static inline void launch_gemm(const float* A, const float* B, float* C,
                               int M, int K, int Nc, hipStream_t stream) {
  dim3 grid((Nc + NTILE - 1) / NTILE, (M + 255) / 256);
  gemm_wmma_f32<<<grid, 128, 0, stream>>>(A, B, C, M, K, Nc);
}

extern "C" void kernel_launch(void* const* d_in, const int* in_sizes, int n_in,
                              void* d_out, int out_size, void* d_ws, size_t ws_size,
                              hipStream_t stream) {
  (void)in_sizes; (void)n_in; (void)out_size; (void)ws_size;
  const float* x     = (const float*)d_in[0];
  const float* graph = (const float*)d_in[1];
  const float* in_w  = (const float*)d_in[2];
  const float* in_b  = (const float*)d_in[3];

  float* ws = (float*)d_ws;
  size_t off = 0;
  float* h    = ws + off; off += (size_t)BATCH * DOUT * TIN0 * NNODE;  // 24.58M
  float* lapT = ws + off; off += (size_t)NNODE * NNODE;                // 4M
  float* c2T  = ws + off; off += (size_t)NNODE * NNODE;                // 4M
  float* deg  = ws + off; off += 2048;
  float* glu  = ws + off; off += (size_t)BATCH * CMID * 10 * NNODE;    // 10.24M
  float* z1   = ws + off; off += (size_t)BATCH * CMID * 10 * NNODE;
  float* z2   = ws + off; off += (size_t)BATCH * CMID * 10 * NNODE;
  float* sc   = ws + off; off += (size_t)BATCH * CMID * 10 * NNODE;
  float* rt   = ws + off; off += (size_t)BATCH * DOUT * 8 * NNODE;     // 16.38M

  const size_t NN = (size_t)NNODE * NNODE;

  // --- Chebyshev basis (transposed): lapT, c2T = (2 lap@lap - I)^T ---
  degree_kernel<<<NNODE, 256, 0, stream>>>(graph, deg);
  lapT_kernel<<<(int)((NN + 255) / 256), 256, 0, stream>>>(graph, deg, lapT);
  launch_gemm(lapT, lapT, c2T, NNODE, NNODE, NNODE, stream);           // lapT @ lapT
  cheb2_fix_kernel<<<(int)((NN + 255) / 256), 256, 0, stream>>>(c2T);

  // --- input 1x1 conv ---
  {
    size_t tot = (size_t)BATCH * DOUT * TIN0 * NNODE;
    input_conv_kernel<<<(int)((tot + 255) / 256), 256, 0, stream>>>(x, in_w, in_b, h);
  }

  // --- two ST blocks ---
  const float* cur_in = h;
  int Tin = TIN0;
  for (int blk = 0; blk < 2; ++blk) {
    const int base = 4 + blk * 10;
    const float* t1_w  = (const float*)d_in[base + 0];
    const float* t1_b  = (const float*)d_in[base + 1];
    const float* t1_aw = (const float*)d_in[base + 2];
    const float* t1_ab = (const float*)d_in[base + 3];
    const float* s_th  = (const float*)d_in[base + 4];
    const float* s_b   = (const float*)d_in[base + 5];
    const float* t2_w  = (const float*)d_in[base + 6];
    const float* t2_b  = (const float*)d_in[base + 7];
    const float* ln_g  = (const float*)d_in[base + 8];
    const float* ln_b  = (const float*)d_in[base + 9];

    const int Tg = Tin - 2;        // after GLU tconv
    const int Tr = Tg - 2;         // after ReLU tconv
    const int M  = BATCH * CMID * Tg;

    {
      size_t tot = (size_t)BATCH * CMID * Tg * NNODE;
      glu_kernel<<<(int)((tot + 255) / 256), 256, 0, stream>>>(
          cur_in, t1_w, t1_b, t1_aw, t1_ab, glu, Tin);
    }
    // graph conv GEMMs (k=1: lapT, k=2: c2T); k=0 handled as identity in sconv_kernel
    launch_gemm(glu, lapT, z1, M, NNODE, NNODE, stream);
    launch_gemm(glu, c2T, z2, M, NNODE, NNODE, stream);
    {
      size_t tot = (size_t)BATCH * CMID * Tg * NNODE;
      sconv_kernel<<<(int)((tot + 255) / 256), 256, 0, stream>>>(
          glu, z1, z2, s_th, s_b, sc, Tg);
    }
    {
      size_t tot = (size_t)BATCH * DOUT * Tr * NNODE;
      relu_tconv_kernel<<<(int)((tot + 255) / 256), 256, 0, stream>>>(
          sc, t2_w, t2_b, rt, Tg);
    }
    float* ln_out = (blk == 0) ? h : (float*)d_out;   // block1 reuses h buffer
    layernorm_kernel<<<BATCH * Tr, 256, 0, stream>>>(rt, ln_g, ln_b, ln_out, Tr);

    cur_in = ln_out;
    Tin = Tr;
  }
}